// GNN_auto_19774029430901
// MI455X (gfx1250) — compile-verified
//
#include <hip/hip_runtime.h>
#include <hip/hip_bf16.h>
#include <math.h>

typedef float v2f __attribute__((ext_vector_type(2)));
typedef float v8f __attribute__((ext_vector_type(8)));

#define D64 64

// ---------------------------------------------------------------------------
// out[M x 64] = act(in[M x 64] @ W^T (+ bias)), W is [64 x 64] row-major.
// 1 wave => one 16-row M tile, all 64 N columns (4 accumulators).
// K=64 done as 16 steps of V_WMMA_F32_16X16X4_F32.
// ---------------------------------------------------------------------------
__global__ __launch_bounds__(256) void gemm64_wmma_kernel(
    const float* __restrict__ in, const float* __restrict__ w,
    const float* __restrict__ bias, float* __restrict__ out,
    int M, int do_relu)
{
    __shared__ float lds[8 * 16 * 68];           // 8 waves x 16x64 tile, stride 68
    const int lane = threadIdx.x & 31;
    const int wid  = threadIdx.x >> 5;
    const int tile = blockIdx.x * 8 + wid;
    const int ntiles = (M + 15) >> 4;
    if (tile >= ntiles) return;                  // wave-uniform: EXEC stays all-1s
    const int row0 = tile * 16;
    float* T = &lds[wid * 16 * 68];

    // Stage A tile (16x64) into LDS, coalesced float4; zero-fill M tail.
    #pragma unroll
    for (int i = 0; i < 8; ++i) {
        int li = i * 32 + lane;
        int r  = li >> 4;
        int c4 = (li & 15) * 4;
        float4 v = make_float4(0.f, 0.f, 0.f, 0.f);
        if (row0 + r < M) v = *(const float4*)&in[(size_t)(row0 + r) * D64 + c4];
        *(float4*)&T[r * 68 + c4] = v;
    }
    __syncthreads();

    const int m  = lane & 15;                    // A row / C column within tile
    const int kg = lane >> 4;                    // lane half selects K pair / M half

    // A fragments for all 16 K-steps: a[s] = A[m][4s + 2*kg + {0,1}]
    v2f a[16];
    #pragma unroll
    for (int s = 0; s < 16; ++s) {
        const float2 t = *(const float2*)&T[m * 68 + 4 * s + 2 * kg];
        a[s].x = t.x; a[s].y = t.y;
    }

    #pragma unroll
    for (int nt = 0; nt < 4; ++nt) {
        const int n = nt * 16 + m;               // this lane's output column
        v8f acc = {};
        #pragma unroll
        for (int s = 0; s < 16; ++s) {
            // B[k][n] = W[n*64 + k]; fragment = W[n][4s + 2*kg + {0,1}]
            const float2 t = *(const float2*)&w[n * D64 + 4 * s + 2 * kg];
            v2f b; b.x = t.x; b.y = t.y;
            acc = __builtin_amdgcn_wmma_f32_16x16x4_f32(
                false, a[s], false, b, (short)0, acc, false, false);
        }
        const float bv = bias ? bias[n] : 0.f;
        #pragma unroll
        for (int i = 0; i < 8; ++i) {
            int row = row0 + 8 * kg + i;         // C/D layout: vgpr i -> M = i + 8*kg
            if (row < M) {
                float v = acc[i] + bv;
                if (do_relu) v = fmaxf(v, 0.f);
                out[(size_t)row * D64 + n] = v;
            }
        }
    }
}

// ---------------------------------------------------------------------------
// hrRaw[r] = emb[mapping[r]]  (r < n_rel);  hq[b] = emb[mapping[q_rel[b]]]
// ---------------------------------------------------------------------------
__global__ void gather_rows_kernel(const float* __restrict__ emb,
                                   const int* __restrict__ mapping,
                                   const int* __restrict__ q_rel,
                                   float* __restrict__ hrRaw,
                                   float* __restrict__ hq,
                                   int n_rel)
{
    int row = blockIdx.x;
    int t   = threadIdx.x;
    if (row < n_rel) {
        hrRaw[(size_t)row * D64 + t] = emb[(size_t)mapping[row] * D64 + t];
    } else {
        int b = row - n_rel;
        hq[(size_t)b * D64 + t] = emb[(size_t)mapping[q_rel[b]] * D64 + t];
    }
}

__global__ void zero_kernel(float* __restrict__ p, int n4)
{
    int i = blockIdx.x * blockDim.x + threadIdx.x;
    if (i < n4) *(float4*)&p[(size_t)i * 4] = make_float4(0.f, 0.f, 0.f, 0.f);
}

// ---------------------------------------------------------------------------
// One wave per edge; lane j owns dims {2j, 2j+1} (256B coalesced row reads).
//   pre  = hsW[sub] + hrW[rel] + qr[r_idx]
//   alpha= sigmoid(relu(pre).walpha + b)      (wave shfl_xor reduction)
//   agg[obj] += alpha * hidden[sub] * hrRaw[rel]   (global_atomic_add_f32)
// Indices are wave-uniform: force them scalar so address math stays in SALU.
// ---------------------------------------------------------------------------
__global__ __launch_bounds__(256) void edge_kernel(
    const float* __restrict__ hidden, const int* __restrict__ edges,
    const int* __restrict__ r_idx,
    const float* __restrict__ hsW, const float* __restrict__ hrW,
    const float* __restrict__ hrRaw, const float* __restrict__ qr,
    const float* __restrict__ walpha_w, const float* __restrict__ walpha_b,
    float* __restrict__ agg, int n_edge)
{
    const int e    = (int)((blockIdx.x * blockDim.x + threadIdx.x) >> 5);
    const int lane = threadIdx.x & 31;
    if (e >= n_edge) return;

    const int sub = __builtin_amdgcn_readfirstlane(edges[e * 3 + 0]);
    const int rel = __builtin_amdgcn_readfirstlane(edges[e * 3 + 1]);
    const int obj = __builtin_amdgcn_readfirstlane(edges[e * 3 + 2]);
    const int r   = __builtin_amdgcn_readfirstlane(r_idx[e]);
    const int c   = lane * 2;

    const float2 ps = *(const float2*)&hsW[(size_t)sub * D64 + c];
    const float2 pr = *(const float2*)&hrW[(size_t)rel * D64 + c];
    const float2 pq = *(const float2*)&qr[(size_t)r * D64 + c];
    const float px = fmaxf(ps.x + pr.x + pq.x, 0.f);
    const float py = fmaxf(ps.y + pr.y + pq.y, 0.f);

    const float2 wa = *(const float2*)&walpha_w[c];
    float s = px * wa.x + py * wa.y;
    #pragma unroll
    for (int off = 16; off > 0; off >>= 1) s += __shfl_xor(s, off, 32);
    const float alpha = 1.f / (1.f + __expf(-(s + walpha_b[0])));

    const float2 hs = *(const float2*)&hidden[(size_t)sub * D64 + c];
    const float2 hr = *(const float2*)&hrRaw[(size_t)rel * D64 + c];
    const float mx = alpha * hs.x * hr.x;
    const float my = alpha * hs.y * hr.y;

    // Hardware FP32 atomics (global_atomic_add_f32), not a CAS loop.
    unsafeAtomicAdd(&agg[(size_t)obj * D64 + c],     mx);
    unsafeAtomicAdd(&agg[(size_t)obj * D64 + c + 1], my);
}

// ---------------------------------------------------------------------------
extern "C" void kernel_launch(void* const* d_in, const int* in_sizes, int n_in,
                              void* d_out, int out_size, void* d_ws, size_t ws_size,
                              hipStream_t stream)
{
    const float* hidden   = (const float*)d_in[0];
    const int*   edges    = (const int*)d_in[1];
    const float* emb      = (const float*)d_in[2];
    const int*   mapping  = (const int*)d_in[3];
    const int*   q_rel    = (const int*)d_in[4];
    const int*   r_idx    = (const int*)d_in[5];
    /* d_in[6] = n_node scalar (device ptr) — sizes used instead */
    const float* Ws_w     = (const float*)d_in[7];
    const float* Wr_w     = (const float*)d_in[8];
    const float* Wqr_w    = (const float*)d_in[9];
    const float* Wqr_b    = (const float*)d_in[10];
    const float* walpha_w = (const float*)d_in[11];
    const float* walpha_b = (const float*)d_in[12];
    const float* Wh_w     = (const float*)d_in[13];
    float* out = (float*)d_out;

    const int n_node = in_sizes[0] / D64;
    const int n_edge = in_sizes[1] / 3;
    const int n_rel  = in_sizes[3];
    const int Bq     = in_sizes[4];

    // Workspace layout (floats): hsW | hrRaw | hrW | hq | qr  (~13.1 MB total)
    float* ws    = (float*)d_ws;
    float* hsW   = ws;
    float* hrRaw = hsW   + (size_t)n_node * D64;
    float* hrW   = hrRaw + (size_t)((n_rel + 15) & ~15) * D64;
    float* hq    = hrW   + (size_t)((n_rel + 15) & ~15) * D64;
    float* qrb   = hq    + (size_t)((Bq + 15) & ~15) * D64;

    auto gemm_blocks = [](int M) { int t = (M + 15) / 16; return (t + 7) / 8; };

    // 1) zero the accumulator (d_out doubles as agg buffer)
    {
        int n4 = n_node * (D64 / 4);
        zero_kernel<<<(n4 + 255) / 256, 256, 0, stream>>>(out, n4);
    }
    // 2) relation / query gathers
    gather_rows_kernel<<<n_rel + Bq, D64, 0, stream>>>(emb, mapping, q_rel,
                                                       hrRaw, hq, n_rel);
    // 3) small per-relation / per-query GEMMs + big per-node GEMM (fp32 WMMA)
    gemm64_wmma_kernel<<<gemm_blocks(n_rel), 256, 0, stream>>>(
        hrRaw, Wr_w, nullptr, hrW, n_rel, 0);
    gemm64_wmma_kernel<<<gemm_blocks(Bq), 256, 0, stream>>>(
        hq, Wqr_w, Wqr_b, qrb, Bq, 0);
    gemm64_wmma_kernel<<<gemm_blocks(n_node), 256, 0, stream>>>(
        hidden, Ws_w, nullptr, hsW, n_node, 0);
    // 4) edge phase: gather + attention + atomic scatter into d_out
    {
        long long threads = (long long)n_edge * 32;
        int blocks = (int)((threads + 255) / 256);
        edge_kernel<<<blocks, 256, 0, stream>>>(hidden, edges, r_idx, hsW, hrW,
                                                hrRaw, qrb, walpha_w, walpha_b,
                                                out, n_edge);
    }
    // 5) final per-node GEMM + ReLU, in-place on d_out (block-local rows only)
    gemm64_wmma_kernel<<<gemm_blocks(n_node), 256, 0, stream>>>(
        out, Wh_w, nullptr, out, n_node, 1);
}